// Quantizer_17815524344475
// MI455X (gfx1250) — compile-verified
//
#include <hip/hip_runtime.h>

typedef __attribute__((ext_vector_type(2))) float v2f;
typedef __attribute__((ext_vector_type(8))) float v8f;
typedef int v4i __attribute__((vector_size(4 * sizeof(int))));

#define HW   4096   // 64*64
#define NC   256    // channels
#define DH   32     // per-head dim
#define KC   256    // codebook entries
#define NH   8      // heads
#define NIMG 8

#define GPTR(p) ((__attribute__((address_space(1))) v4i*)(p))
#define LPTR(p) ((__attribute__((address_space(3))) v4i*)(p))

// ---------------------------------------------------------------------------
// Prep: A[m] = (codebook[m] @ wk[m]^T) @ wq[m]   [KC,DH]
//       Vp[m] = codebook[m] @ wv[m]^T            [KC,DH]
// ---------------------------------------------------------------------------
__global__ void quant_prep(const float* __restrict__ codebook,
                           const float* __restrict__ wq,
                           const float* __restrict__ wk,
                           const float* __restrict__ wv,
                           float* __restrict__ ws) {
  __shared__ float swq[DH * DH];
  __shared__ float swk[DH * DH];
  __shared__ float swv[DH * DH];
  const int m = blockIdx.x;
  const int t = threadIdx.x;            // 0..255 -> one codebook row each
  for (int i = t; i < DH * DH; i += blockDim.x) {
    swq[i] = wq[m * DH * DH + i];
    swk[i] = wk[m * DH * DH + i];
    swv[i] = wv[m * DH * DH + i];
  }
  __syncthreads();

  const int kk = t;
  float cb[DH];
  const float* crow = codebook + ((size_t)m * KC + kk) * DH;
#pragma unroll
  for (int d = 0; d < DH; ++d) cb[d] = crow[d];

  float kp[DH];
#pragma unroll
  for (int c = 0; c < DH; ++c) {
    float s = 0.f;
#pragma unroll
    for (int d = 0; d < DH; ++d) s += cb[d] * swk[c * DH + d];
    kp[c] = s;
  }

  float* arow = ws + (size_t)m * KC * DH + (size_t)kk * DH;
  float* vrow = ws + (size_t)NH * KC * DH + (size_t)m * KC * DH + (size_t)kk * DH;
#pragma unroll
  for (int d = 0; d < DH; ++d) {
    float s = 0.f;
#pragma unroll
    for (int c = 0; c < DH; ++c) s += kp[c] * swq[c * DH + d];
    arow[d] = s;
  }
#pragma unroll
  for (int c = 0; c < DH; ++c) {
    float s = 0.f;
#pragma unroll
    for (int d = 0; d < DH; ++d) s += cb[d] * swv[c * DH + d];
    vrow[c] = s;
  }
}

// ---------------------------------------------------------------------------
// Main: per (64-pixel tile, head) block, 4 wave32s.
//   GEMM1: logit[16p x 256k] = X[16p x 32d] * A^T       (f32 WMMA, K=4 steps)
//   GEMM2: quant[16p x 32c]  = logit[16p x 256k] * Vp   (f32 WMMA)
// All bulk global I/O of results goes LDS <-> global via the async engine.
// ---------------------------------------------------------------------------
__global__ void quant_main(const float* __restrict__ latent,
                           const float* __restrict__ Amat,   // [NH][KC][DH]
                           const float* __restrict__ Vmat,   // [NH][KC][DH]
                           float* __restrict__ qout,
                           float* __restrict__ codeout,
                           float* __restrict__ logitout) {
  extern __shared__ float smem[];
  float* sA   = smem;                 // 8192 floats
  float* sV   = smem + 8192;          // 8192 floats
  float* sLog = smem + 16384;         // 4 waves * 16p * 256k = 16384 floats
  float* sQ   = smem + 32768;         // 4 waves * 32c * 16p  = 2048 floats

  const int tid  = threadIdx.x;       // 0..127 (4 wave32s)
  const int lane = tid & 31;
  const int wvid = tid >> 5;          // 0..3
  const int r    = lane & 15;         // row-in-group / N index
  const int g    = lane >> 4;         // lane-group (K split per ISA layout)

  const int tile = blockIdx.x;        // 0..511
  const int m    = blockIdx.y;        // head
  const int gp0  = tile * 64;         // global pixel base (n*4096 + p)
  const int nimg = gp0 >> 12;
  const int p0   = gp0 & (HW - 1);
  const int pbase = p0 + wvid * 16;   // this wave's 16 pixels

  // --- stage per-head A and Vp into LDS via async global->LDS (ASYNCcnt) ---
  {
    const float4* gA = (const float4*)(Amat + (size_t)m * KC * DH);
    const float4* gV = (const float4*)(Vmat + (size_t)m * KC * DH);
    float4* lA = (float4*)sA;
    float4* lV = (float4*)sV;
    for (int i = tid; i < (KC * DH) / 4; i += (int)blockDim.x) {
      __builtin_amdgcn_global_load_async_to_lds_b128(GPTR(gA + i), LPTR(lA + i), 0, 0);
      __builtin_amdgcn_global_load_async_to_lds_b128(GPTR(gV + i), LPTR(lV + i), 0, 0);
    }
  }

  // --- load X fragments (overlaps with async staging) ----------------------
  const float* latB = latent + (size_t)nimg * NC * HW + (size_t)m * DH * HW;
  __builtin_prefetch(latB + pbase, 0, 1);   // global_prefetch_b8 path
  v2f afrag[8];
#pragma unroll
  for (int k4 = 0; k4 < 8; ++k4) {
    // lane holds X[pixel = r][d = 4*k4 + 2*g + {0,1}]; d-stride = HW floats
    const float* ap = latB + (size_t)(k4 * 4 + 2 * g) * HW + pbase + r;
    afrag[k4].x = ap[0];
    afrag[k4].y = ap[HW];
  }

  __builtin_amdgcn_s_wait_asynccnt(0);
  __syncthreads();

  // --- GEMM1: 16 output N-tiles over kk, 8 K-steps each ---------------------
  for (int kt = 0; kt < 16; ++kt) {
    v8f acc = {};
#pragma unroll
    for (int k4 = 0; k4 < 8; ++k4) {
      v2f b;  // B[dlocal][n] = A[kt*16+n][4*k4+dlocal]
      b.x = sA[(kt * 16 + r) * DH + k4 * 4 + 2 * g];
      b.y = sA[(kt * 16 + r) * DH + k4 * 4 + 2 * g + 1];
      acc = __builtin_amdgcn_wmma_f32_16x16x4_f32(
          false, afrag[k4], false, b, (short)0, acc, false, false);
    }
    // C/D layout: VGPR rr, lane-group g -> pixel rr+8g, kk = kt*16 + r
#pragma unroll
    for (int rr = 0; rr < 8; ++rr) {
      const int pl = rr + 8 * g;
      sLog[wvid * 4096 + pl * KC + kt * 16 + r] = acc[rr];
    }
  }

  // --- flush this wave's 16KB logit block LDS -> global (async b128) -------
  asm volatile("s_wait_dscnt 0" ::: "memory");  // LDS writes visible to async engine
  {
    float* logW = logitout + (((size_t)(nimg * HW + pbase)) * NH + m) * KC;
#pragma unroll 4
    for (int i = 0; i < 32; ++i) {
      const int f4 = i * 32 + lane;       // 0..1023 float4s for this wave
      const int pl = f4 >> 6;             // pixel 0..15
      const int kq = (f4 & 63) * 4;       // kk within row
      __builtin_amdgcn_global_store_async_from_lds_b128(
          GPTR(logW + (size_t)pl * NH * KC + kq),
          LPTR(sLog + wvid * 4096 + pl * KC + kq), 0, 0);
    }
  }

  __syncthreads();

  // --- argmax (waves 0,1 cover the block's 64 pixels; wave-uniform branch) --
  if (tid < 64) {
    const float* lp = sLog + (tid >> 4) * 4096 + (tid & 15) * KC;
    float best = lp[0];
    int bi = 0;
    for (int kk = 1; kk < KC; ++kk) {
      const float v = lp[kk];
      if (v > best) { best = v; bi = kk; }
    }
    codeout[(size_t)(gp0 + tid) * NH + m] = (float)bi;
  }

  // --- GEMM2: quant[16p x 32c], K = 256 in 64 steps -------------------------
  const float* myLog = sLog + wvid * 4096;
#pragma unroll
  for (int ct = 0; ct < 2; ++ct) {
    v8f qacc = {};
    for (int k4 = 0; k4 < 64; ++k4) {
      v2f a2;  // A[pixel = r][kk = 4*k4 + 2*g + {0,1}]
      a2.x = myLog[r * KC + k4 * 4 + 2 * g];
      a2.y = myLog[r * KC + k4 * 4 + 2 * g + 1];
      v2f b2;  // B[kklocal][c] = Vp[4*k4 + 2*g + {0,1}][ct*16 + c]
      b2.x = sV[(k4 * 4 + 2 * g) * DH + ct * 16 + r];
      b2.y = sV[(k4 * 4 + 2 * g + 1) * DH + ct * 16 + r];
      qacc = __builtin_amdgcn_wmma_f32_16x16x4_f32(
          false, a2, false, b2, (short)0, qacc, false, false);
    }
    // park transposed in LDS: sQ[c][pixel] for channel-major async store
#pragma unroll
    for (int rr = 0; rr < 8; ++rr) {
      sQ[wvid * 512 + (ct * 16 + r) * 16 + rr + 8 * g] = qacc[rr];
    }
  }
  __syncthreads();                              // drains DScnt; sQ complete

  // --- quantized store [n, m*32+c, hw]: async LDS -> global b128 ------------
  {
    float* qB = qout + (size_t)nimg * NC * HW + (size_t)m * DH * HW + p0;
#pragma unroll
    for (int i = 0; i < 4; ++i) {
      const int f4   = i * 128 + tid;     // 0..511 float4s
      const int wreg = f4 >> 7;           // source wave region
      const int rem  = f4 & 127;
      const int c    = rem >> 2;
      const int pp   = (rem & 3) * 4;
      __builtin_amdgcn_global_store_async_from_lds_b128(
          GPTR(qB + (size_t)c * HW + wreg * 16 + pp),
          LPTR(sQ + wreg * 512 + c * 16 + pp), 0, 0);
    }
  }
  // s_endpgm performs an implicit wait-idle, draining ASYNCcnt.
}

// ---------------------------------------------------------------------------
extern "C" void kernel_launch(void* const* d_in, const int* in_sizes, int n_in,
                              void* d_out, int out_size, void* d_ws, size_t ws_size,
                              hipStream_t stream) {
  (void)in_sizes; (void)n_in; (void)out_size; (void)ws_size;
  const float* latent   = (const float*)d_in[0];
  const float* codebook = (const float*)d_in[1];
  const float* wq       = (const float*)d_in[2];
  const float* wk       = (const float*)d_in[3];
  const float* wv       = (const float*)d_in[4];

  float* ws   = (float*)d_ws;
  float* Amat = ws;                                   // 8*256*32 floats
  float* Vmat = ws + (size_t)NH * KC * DH;            // 8*256*32 floats

  float* qout     = (float*)d_out;                    // 8*256*4096
  float* codeout  = qout + (size_t)NIMG * NC * HW;    // 8*4096*8
  float* logitout = codeout + (size_t)NIMG * HW * NH; // 8*4096*8*256

  quant_prep<<<dim3(NH), dim3(256), 0, stream>>>(codebook, wq, wk, wv, ws);

  const size_t smem_bytes = (size_t)(8192 + 8192 + 16384 + 2048) * sizeof(float);
  quant_main<<<dim3(512, NH), dim3(128), smem_bytes, stream>>>(
      latent, Amat, Vmat, qout, codeout, logitout);
}